// Modular_86827058856242
// MI455X (gfx1250) — compile-verified
//
#include <hip/hip_runtime.h>
#include <hip/hip_bf16.h>

#define BB   32768
#define EE   256
#define DD   512
#define RR   8
#define OPD  16
#define XDIM 18     // 2 + OP
#define SIN  530    // D + OP + 2

typedef float v2f __attribute__((ext_vector_type(2)));
typedef float v8f __attribute__((ext_vector_type(8)));

// ---------------------------------------------------------------------------
// Stage 1: first layers of the three encoders (rank-1 / K=16) -> H[B,768]
//   cols [0,256)   : relu(x0 * dW1 + db1)
//   cols [256,512) : relu(x1 * dW1 + db1)
//   cols [512,768) : relu(x[2:18] @ oW1 + ob1)
// ---------------------------------------------------------------------------
__global__ void hidden3_kernel(const float* __restrict__ x,
                               const float* __restrict__ dW1,
                               const float* __restrict__ db1,
                               const float* __restrict__ oW1,
                               const float* __restrict__ ob1,
                               float* __restrict__ H)
{
    int idx = blockIdx.x * blockDim.x + threadIdx.x; // b*768 + c
    int b = idx / 768;
    int c = idx - b * 768;
    if (b >= BB) return;
    const float* xb = x + (size_t)b * XDIM;
    float v;
    if (c < 512) {
        int e = c & 255;
        float xin = xb[c >> 8];
        v = fmaf(xin, dW1[e], db1[e]);
    } else {
        int e = c - 512;
        v = ob1[e];
#pragma unroll
        for (int k = 0; k < OPD; ++k) v = fmaf(xb[2 + k], oW1[k * EE + e], v);
    }
    H[(size_t)idx] = fmaxf(v, 0.f);
}

// ---------------------------------------------------------------------------
// Pack g1w [R,768,64] -> Wg1 [768, 512] row-major (n = r*64+e)
// ---------------------------------------------------------------------------
__global__ void pack_g1_kernel(const float* __restrict__ g1w,
                               float* __restrict__ Wg1)
{
    int idx = blockIdx.x * blockDim.x + threadIdx.x; // k*512 + n
    if (idx >= 768 * 512) return;
    int k = idx >> 9;
    int n = idx & 511;
    Wg1[idx] = g1w[(size_t)(n >> 6) * (768 * 64) + k * 64 + (n & 63)];
}

// ---------------------------------------------------------------------------
// Generic fp32 WMMA GEMM:  C[M,N] = act(A[M,K] @ W[K,N] + bias[N])
//   block = 256 threads (8 waves). Tile: 128 rows x 64 cols.
//   Wave w owns rows [w*16, w*16+16), 4 accumulator tiles along N.
//   K chunked by 16, A(128x16)/B(16x64) staged in LDS.
//   Requires: M%128==0, N%64==0, K%16==0, 16B-aligned rows (all call sites ok)
// ---------------------------------------------------------------------------
__launch_bounds__(256)
__global__ void gemm_f32_kernel(const float* __restrict__ A, int lda,
                                const float* __restrict__ W, int ldw,
                                const float* __restrict__ bias,
                                float* __restrict__ C, int ldc,
                                int K, int relu)
{
    __shared__ float As[128][20]; // stride 20: 16B-aligned float4 stores, conflict-free reads
    __shared__ float Bs[16][64];

    const int tid  = threadIdx.x;
    const int wave = tid >> 5;
    const int lane = tid & 31;
    const int lm   = lane & 15;
    const int hi   = lane >> 4;
    const int rb   = blockIdx.x * 128;
    const int cb   = blockIdx.y * 64;

    v8f acc[4] = {};

    const int arow  = tid >> 1;        // 0..127
    const int acol8 = (tid & 1) * 8;   // 0 or 8
    const int bkk   = tid >> 4;        // 0..15
    const int bn4   = (tid & 15) * 4;  // 0..60

    for (int k0 = 0; k0 < K; k0 += 16) {
        __syncthreads();
        // cooperative A tile load (rows contiguous, float4)
        const float* ap = A + (size_t)(rb + arow) * lda + k0 + acol8;
        *(float4*)&As[arow][acol8]     = *(const float4*)(ap);
        *(float4*)&As[arow][acol8 + 4] = *(const float4*)(ap + 4);
        // cooperative B tile load
        const float* wp = W + (size_t)(k0 + bkk) * ldw + cb + bn4;
        *(float4*)&Bs[bkk][bn4] = *(const float4*)(wp);
        __syncthreads();

#pragma unroll
        for (int kk = 0; kk < 16; kk += 4) {
            v2f a;
            a.x = As[wave * 16 + lm][kk + 2 * hi];
            a.y = As[wave * 16 + lm][kk + 2 * hi + 1];
#pragma unroll
            for (int t = 0; t < 4; ++t) {
                v2f b;
                b.x = Bs[kk + 2 * hi][t * 16 + lm];
                b.y = Bs[kk + 2 * hi + 1][t * 16 + lm];
                acc[t] = __builtin_amdgcn_wmma_f32_16x16x4_f32(
                    false, a, false, b, (short)0, acc[t], false, false);
            }
        }
    }

    // epilogue: C/D layout -> element (M = v + 8*hi, N = t*16 + lm)
#pragma unroll
    for (int t = 0; t < 4; ++t) {
        int col = cb + t * 16 + lm;
        float bv = bias[col];
#pragma unroll
        for (int v = 0; v < 8; ++v) {
            int row = rb + wave * 16 + v + 8 * hi;
            float val = acc[t][v] + bv;
            if (relu) val = fmaxf(val, 0.f);
            C[(size_t)row * ldc + col] = val;
        }
    }
}

// ---------------------------------------------------------------------------
// Scorer (dominant stage, ~52% of FLOPs): rows are (b*8 + r), M = B*R = 262144
//   hidden = relu(concat(out[b,r,:], x[b,:]) @ sW1 + sb1)   (K = 530, N = 256)
//   score_raw = hidden . sW2 + sb2
// Block = 256 threads / 8 waves / 128 rows. K chunked by 16 with LDS staging:
//   As: gathered 128x16 tile of concat(out,x) (zero-padded K tail)
//   Bs: 16x256 tile of sW1 shared by all 8 waves (8x less global traffic)
// Wave holds all 256 hidden dims in 16 fp32-WMMA accumulators, then fuses the
// ReLU + sW2 dot with shfl_xor reductions inside 16-lane halves.
// ---------------------------------------------------------------------------
__launch_bounds__(256)
__global__ void scorer_kernel(const float* __restrict__ outbuf,
                              const float* __restrict__ x,
                              const float* __restrict__ sW1,
                              const float* __restrict__ sb1,
                              const float* __restrict__ sW2,
                              const float* __restrict__ sb2,
                              float* __restrict__ score_raw)
{
    __shared__ float As[128][20];  // padded stride: aligned float4 stores, conflict-free reads
    __shared__ float Bs[16][256];

    const int tid  = threadIdx.x;
    const int wave = tid >> 5;
    const int lane = tid & 31;
    const int lm   = lane & 15;
    const int hi   = lane >> 4;
    const int rowbase = blockIdx.x * 128;

    // cooperative A-tile load indices: thread -> (row, 8-col half)
    const int trow  = tid >> 1;          // 0..127
    const int tcol8 = (tid & 1) * 8;     // 0 or 8
    const int grow  = rowbase + trow;
    const float* arow_out = outbuf + (size_t)(grow >> 3) * (RR * DD) + (size_t)(grow & 7) * DD;
    const float* arow_x   = x + (size_t)(grow >> 3) * XDIM;

    // cooperative B-tile load indices: thread -> (k-row, 16-col slab)
    const int bkk  = tid >> 4;           // 0..15
    const int bn16 = (tid & 15) * 16;    // 0..240

    v8f acc[16] = {};

    for (int k0 = 0; k0 < SIN; k0 += 16) {   // 34 chunks (K padded to 544 with zeros)
        __syncthreads();
        // ---- stage A tile ----
        if (k0 + 16 <= DD) {                 // fully inside the 'out' region: float4 path
            const float* ap = arow_out + k0 + tcol8;
            *(float4*)&As[trow][tcol8]     = *(const float4*)(ap);
            *(float4*)&As[trow][tcol8 + 4] = *(const float4*)(ap + 4);
        } else {                             // last two chunks: guarded out|x|0 gather
#pragma unroll
            for (int e = 0; e < 8; ++e) {
                int k = k0 + tcol8 + e;
                float v = 0.f;
                if (k < DD)       v = arow_out[k];
                else if (k < SIN) v = arow_x[k - DD];
                As[trow][tcol8 + e] = v;
            }
        }
        // ---- stage B tile (sW1), zero-fill padded K rows ----
        {
            int kb = k0 + bkk;
            if (kb < SIN) {
                const float* wp = sW1 + (size_t)kb * EE + bn16;
#pragma unroll
                for (int q = 0; q < 4; ++q)
                    *(float4*)&Bs[bkk][bn16 + 4 * q] = *(const float4*)(wp + 4 * q);
            } else {
#pragma unroll
                for (int e = 0; e < 16; ++e) Bs[bkk][bn16 + e] = 0.f;
            }
        }
        __syncthreads();

        // ---- 64 unguarded WMMAs per chunk, fragments from LDS ----
#pragma unroll
        for (int kk = 0; kk < 16; kk += 4) {
            v2f a;
            a.x = As[wave * 16 + lm][kk + 2 * hi];
            a.y = As[wave * 16 + lm][kk + 2 * hi + 1];
#pragma unroll
            for (int t = 0; t < 16; ++t) {
                v2f b;
                b.x = Bs[kk + 2 * hi][t * 16 + lm];
                b.y = Bs[kk + 2 * hi + 1][t * 16 + lm];
                acc[t] = __builtin_amdgcn_wmma_f32_16x16x4_f32(
                    false, a, false, b, (short)0, acc[t], false, false);
            }
        }
    }

    const float w2b = sb2[0];
#pragma unroll
    for (int v = 0; v < 8; ++v) {
        float p = 0.f;
#pragma unroll
        for (int t = 0; t < 16; ++t) {
            int n = t * 16 + lm;
            float hv = fmaxf(acc[t][v] + sb1[n], 0.f);
            p = fmaf(hv, sW2[n], p);
        }
        // lanes 0..15 hold row (base+v), lanes 16..31 hold row (base+v+8)
        p += __shfl_xor(p, 1, 32);
        p += __shfl_xor(p, 2, 32);
        p += __shfl_xor(p, 4, 32);
        p += __shfl_xor(p, 8, 32);
        if (lm == 0) score_raw[rowbase + wave * 16 + v + 8 * hi] = p + w2b;
    }
}

// ---------------------------------------------------------------------------
// Softmax over R, emit score output, weighted pooling -> pooled[B,512]
// One wave per sample.
// ---------------------------------------------------------------------------
__global__ void softmax_pool_kernel(const float* __restrict__ score_raw,
                                    const float* __restrict__ outbuf,
                                    float* __restrict__ score_out,
                                    float* __restrict__ pooled)
{
    int wave = threadIdx.x >> 5, lane = threadIdx.x & 31;
    int b = blockIdx.x * 8 + wave;
    if (b >= BB) return;
    const float* sr = score_raw + (size_t)b * RR;
    float s[RR], m = -1e30f;
#pragma unroll
    for (int r = 0; r < RR; ++r) { s[r] = sr[r]; m = fmaxf(m, s[r]); }
    float sum = 0.f;
#pragma unroll
    for (int r = 0; r < RR; ++r) { s[r] = expf(s[r] - m); sum += s[r]; }
    float inv = 1.f / sum;
#pragma unroll
    for (int r = 0; r < RR; ++r) s[r] *= inv;
    if (lane < RR) score_out[(size_t)b * RR + lane] = s[lane];
    const float* ob = outbuf + (size_t)b * (RR * DD);
    for (int d = lane; d < DD; d += 32) {
        float acc = 0.f;
#pragma unroll
        for (int r = 0; r < RR; ++r) acc = fmaf(ob[r * DD + d], s[r], acc);
        pooled[(size_t)b * DD + d] = acc;
    }
}

// ---------------------------------------------------------------------------
// Final decoder dot: dec[b] = decH[b,:] . decW2 + decb2   (one wave / sample)
// ---------------------------------------------------------------------------
__global__ void dec_final_kernel(const float* __restrict__ decH,
                                 const float* __restrict__ decW2,
                                 const float* __restrict__ decb2,
                                 float* __restrict__ dec)
{
    int wave = threadIdx.x >> 5, lane = threadIdx.x & 31;
    int b = blockIdx.x * 8 + wave;
    if (b >= BB) return;
    const float* h = decH + (size_t)b * EE;
    float p = 0.f;
#pragma unroll
    for (int i = 0; i < EE / 32; ++i)
        p = fmaf(h[lane + i * 32], decW2[lane + i * 32], p);
#pragma unroll
    for (int msk = 16; msk; msk >>= 1) p += __shfl_xor(p, msk, 32);
    if (lane == 0) dec[b] = p + decb2[0];
}

// ---------------------------------------------------------------------------
extern "C" void kernel_launch(void* const* d_in, const int* in_sizes, int n_in,
                              void* d_out, int out_size, void* d_ws, size_t ws_size,
                              hipStream_t stream)
{
    const float* x     = (const float*)d_in[0];
    const float* dW1   = (const float*)d_in[1];
    const float* db1   = (const float*)d_in[2];
    const float* dW2   = (const float*)d_in[3];
    const float* db2   = (const float*)d_in[4];
    const float* oW1   = (const float*)d_in[5];
    const float* ob1   = (const float*)d_in[6];
    const float* oW2   = (const float*)d_in[7];
    const float* ob2   = (const float*)d_in[8];
    const float* g1w   = (const float*)d_in[9];
    const float* g1b   = (const float*)d_in[10];
    const float* g2w   = (const float*)d_in[11];
    const float* g2b   = (const float*)d_in[12];
    const float* sW1   = (const float*)d_in[13];
    const float* sb1   = (const float*)d_in[14];
    const float* sW2   = (const float*)d_in[15];
    const float* sb2   = (const float*)d_in[16];
    const float* decW1 = (const float*)d_in[17];
    const float* decb1 = (const float*)d_in[18];
    const float* decW2 = (const float*)d_in[19];
    const float* decb2 = (const float*)d_in[20];

    // workspace layout (floats), ~707 MB, with H-region reuse
    float* ws        = (float*)d_ws;
    float* sample    = ws;                                   // B*768
    float* outbuf    = sample + (size_t)BB * 768;            // B*4096
    float* hbuf      = outbuf + (size_t)BB * 4096;           // B*768 region (reused)
    float* hmid      = hbuf;                                 // B*512 (group-linear-1 out)
    float* pooled    = hbuf;                                 // B*512 (after hmid dead)
    float* decH      = hbuf + (size_t)BB * 512;              // B*256
    float* Wg1       = hbuf + (size_t)BB * 768;              // 768*512
    float* score_raw = Wg1 + 768 * 512;                      // B*8

    float* dec_out   = (float*)d_out;       // [B]
    float* score_out = dec_out + BB;        // [B,8]

    dim3 blk(256);

    hidden3_kernel<<<dim3((BB * 768) / 256), blk, 0, stream>>>(x, dW1, db1, oW1, ob1, hbuf);
    pack_g1_kernel<<<dim3((768 * 512) / 256), blk, 0, stream>>>(g1w, Wg1);

    // encoders' second layers -> sample[B,768]
    gemm_f32_kernel<<<dim3(BB / 128, 4), blk, 0, stream>>>(hbuf,       768, dW2, 256, db2, sample,       768, 256, 0);
    gemm_f32_kernel<<<dim3(BB / 128, 4), blk, 0, stream>>>(hbuf + 256, 768, dW2, 256, db2, sample + 256, 768, 256, 0);
    gemm_f32_kernel<<<dim3(BB / 128, 4), blk, 0, stream>>>(hbuf + 512, 768, oW2, 256, ob2, sample + 512, 768, 256, 0);

    // group-linear 1: h = relu(sample @ Wg1 + g1b)  [B,512]
    gemm_f32_kernel<<<dim3(BB / 128, 8), blk, 0, stream>>>(sample, 768, Wg1, 512, g1b, hmid, 512, 768, 1);

    // group-linear 2 per rule: out[b,r,:] = h[b,r,:] @ g2w[r] + g2b[r]
    for (int r = 0; r < RR; ++r)
        gemm_f32_kernel<<<dim3(BB / 128, 8), blk, 0, stream>>>(
            hmid + r * 64, 512, g2w + (size_t)r * 64 * 512, 512,
            g2b + r * 512, outbuf + r * 512, RR * DD, 64, 0);

    // scorer (WMMA, dominant stage)
    scorer_kernel<<<dim3((BB * RR) / 128), blk, 0, stream>>>(outbuf, x, sW1, sb1, sW2, sb2, score_raw);

    // softmax + pooling (also emits score outputs)
    softmax_pool_kernel<<<dim3(BB / 8), blk, 0, stream>>>(score_raw, outbuf, score_out, pooled);

    // decoder hidden layer (WMMA) + final dot
    gemm_f32_kernel<<<dim3(BB / 128, 4), blk, 0, stream>>>(pooled, 512, decW1, 256, decb1, decH, 256, 512, 1);
    dec_final_kernel<<<dim3(BB / 8), blk, 0, stream>>>(decH, decW2, decb2, dec_out);
}